// SoftLabelLoss_52089363366642
// MI455X (gfx1250) — compile-verified
//
#include <hip/hip_runtime.h>
#include <stdint.h>

// SoftLabelLoss (KL with smoothed soft targets), B=524288 rows, C=170 classes.
// Memory-bound: 356MB logits read once (~15.3us roofline @ 23.3TB/s HBM).
// No matmul structure -> WMMA inapplicable; CDNA5 paths used instead:
//   * global_load_async_to_lds_b64 (ASYNCcnt) double-buffered row streaming,
//     tagged th:TH_LOAD_NT so the one-shot 356MB stream does not evict the
//     hot 113KB sim_matrix / tables from the 192MB L2,
//   * split wait counters (s_wait_asynccnt / s_wait_dscnt),
//   * wave32 shuffle reductions.
// Deterministic two-stage reduction (no float atomics) for graph replays.

#define IGNORE_INDEX (-100)
#define CC 170          // classes (fixed by reference)
#define TPB 256         // 8 waves per block (wave32)
#define WAVES 8
#define NBLK 1024       // B/NBLK = 512 rows per block, 64 rows per wave
#define TILE_PAD 172    // 170 floats padded to 8B multiple

// CDNA5 async ops via inline asm (portable across both toolchains).
// VDST = LDS byte offset (workgroup-relative), VADDR = 64-bit global address.
// Row pitch is 680B (8B- but not 16B-aligned), so b64 chunks are the widest
// always-aligned granule: 85 chunks = lanes 0..31, 0..31, 0..20 => exactly
// 3 async instructions per row (ASYNCcnt += 3).
#define ASYNC_LD_B64_NT(ldsoff, gptr)                                       \
  asm volatile("global_load_async_to_lds_b64 %0, %1, off th:TH_LOAD_NT"     \
               :: "v"(ldsoff), "v"(gptr) : "memory")
#define WAIT_ASYNCCNT(n) asm volatile("s_wait_asynccnt " #n ::: "memory")
#define WAIT_DSCNT0()    asm volatile("s_wait_dscnt 0" ::: "memory")

__device__ __forceinline__ float wsum(float v) {
#pragma unroll
  for (int o = 16; o > 0; o >>= 1) v += __shfl_xor(v, o, 32);
  return v;
}
__device__ __forceinline__ float wmax(float v) {
#pragma unroll
  for (int o = 16; o > 0; o >>= 1) v = fmaxf(v, __shfl_xor(v, o, 32));
  return v;
}

// ---------------- Kernel 1: per-class tables S_c = sum(sim[c]),
// E_c = sum_j soft_j*log(soft_j), soft = 0.8*onehot + 0.2*sim/S. -------------
__global__ void sll_prep(const float* __restrict__ sim,
                         float* __restrict__ Sv, float* __restrict__ Ev) {
  const int c = blockIdx.x * (TPB / 32) + (threadIdx.x >> 5);
  const int L = threadIdx.x & 31;
  if (c >= CC) return;
  const float* row = sim + (size_t)c * CC;
  float s = 0.f;
#pragma unroll
  for (int k = 0; k < 6; ++k) { int j = L + 32 * k; if (j < CC) s += row[j]; }
  s = wsum(s);
  const float inv = 1.0f / s;
  float e = 0.f;
#pragma unroll
  for (int k = 0; k < 6; ++k) {
    int j = L + 32 * k;
    if (j < CC) {
      float r = 0.2f * row[j] * inv + ((j == c) ? 0.8f : 0.f);
      if (r > 0.f) e += r * __logf(r);   // 0*log(0) := 0
    }
  }
  e = wsum(e);
  if (L == 0) { Sv[c] = s; Ev[c] = e; }
}

// ---------------- Kernel 2: main streaming pass over logits. ----------------
// kl_b = E_t - 0.8*l_t - (0.2/S_t)*dot(sim[t], l) + logZ_b, summed per block.
__global__ void __launch_bounds__(TPB)
sll_main(const float* __restrict__ logits, const int* __restrict__ targets,
         const float* __restrict__ sim, const float* __restrict__ Sv,
         const float* __restrict__ Ev, float* __restrict__ partials,
         int rows_per_block) {
  __shared__ float tile[WAVES][2][TILE_PAD];
  __shared__ float sS[CC], sE[CC];
  __shared__ float wacc[WAVES];

  for (int i = threadIdx.x; i < CC; i += TPB) { sS[i] = Sv[i]; sE[i] = Ev[i]; }
  __syncthreads();

  const int w = threadIdx.x >> 5, L = threadIdx.x & 31;
  const int iters = rows_per_block >> 3;                 // rows per wave
  const long long row0 = (long long)blockIdx.x * rows_per_block + w;

  const unsigned lb0 = (unsigned)(uintptr_t)&tile[w][0][0];
  const unsigned lb1 = (unsigned)(uintptr_t)&tile[w][1][0];

  { // prologue: stage first row into buffer 0
    const float* g = logits + (size_t)row0 * CC;
    ASYNC_LD_B64_NT(lb0 + 8u * L,          g + 2 * L);
    ASYNC_LD_B64_NT(lb0 + 256u + 8u * L,   g + 64 + 2 * L);
    if (L < 21) ASYNC_LD_B64_NT(lb0 + 512u + 8u * L, g + 128 + 2 * L);
  }

  float acc = 0.f;
  for (int i = 0; i < iters; ++i) {
    const int cur = i & 1;
    if (i + 1 < iters) {
      const unsigned lb = (cur == 0) ? lb1 : lb0;
      const float* g = logits + (size_t)(row0 + (long long)(i + 1) * WAVES) * CC;
      WAIT_DSCNT0();      // all prior DS reads of the target buffer complete
      ASYNC_LD_B64_NT(lb + 8u * L,          g + 2 * L);
      ASYNC_LD_B64_NT(lb + 256u + 8u * L,   g + 64 + 2 * L);
      if (L < 21) ASYNC_LD_B64_NT(lb + 512u + 8u * L, g + 128 + 2 * L);
      WAIT_ASYNCCNT(3);   // async loads complete in order: current tile ready
    } else {
      WAIT_ASYNCCNT(0);
    }

    const long long row = row0 + (long long)i * WAVES;
    const int t  = targets[row];
    const int tt = (t == IGNORE_INDEX) ? 0 : t;

    const float2* tp = (const float2*)&tile[w][cur][0];
    const float2* sp = (const float2*)(sim + (size_t)tt * CC); // L2-resident
    float2 l0 = tp[L],       l1 = tp[32 + L];
    float2 s0 = sp[L],       s1 = sp[32 + L];
    float2 l2 = make_float2(-INFINITY, -INFINITY);
    float2 s2 = make_float2(0.f, 0.f);
    const bool v2 = (L < 21);
    if (v2) { l2 = tp[64 + L]; s2 = sp[64 + L]; }

    float m = fmaxf(fmaxf(fmaxf(l0.x, l0.y), fmaxf(l1.x, l1.y)),
                    fmaxf(l2.x, l2.y));
    m = wmax(m);

    float z   = __expf(l0.x - m) + __expf(l0.y - m) +
                __expf(l1.x - m) + __expf(l1.y - m);
    float dot = s0.x * l0.x + s0.y * l0.y + s1.x * l1.x + s1.y * l1.y;
    const int j0 = 2 * L, j1 = 64 + 2 * L, j2 = 128 + 2 * L;
    float lt = 0.f;
    lt += (j0 == tt) ? l0.x : 0.f;  lt += (j0 + 1 == tt) ? l0.y : 0.f;
    lt += (j1 == tt) ? l1.x : 0.f;  lt += (j1 + 1 == tt) ? l1.y : 0.f;
    if (v2) {
      z   += __expf(l2.x - m) + __expf(l2.y - m);
      dot += s2.x * l2.x + s2.y * l2.y;
      lt  += (j2 == tt) ? l2.x : 0.f;  lt += (j2 + 1 == tt) ? l2.y : 0.f;
    }
    z = wsum(z); dot = wsum(dot); lt = wsum(lt);

    if (L == 0) {
      const float logZ = m + __logf(z);
      float kl = 0.f;
      if (t != IGNORE_INDEX)
        kl = sE[tt] - 0.8f * lt - 0.2f * dot / sS[tt] + logZ;
      acc += kl;
    }
  }

  if (L == 0) wacc[w] = acc;
  __syncthreads();
  if (threadIdx.x == 0) {
    float s = 0.f;
#pragma unroll
    for (int k = 0; k < WAVES; ++k) s += wacc[k];
    partials[blockIdx.x] = s;
  }
}

// ---------------- Kernel 3: deterministic final reduction. ------------------
__global__ void sll_finish(const float* __restrict__ partials, int n,
                           float* __restrict__ out, float Bf) {
  __shared__ float red[TPB];
  float s = 0.f;
  for (int i = threadIdx.x; i < n; i += TPB) s += partials[i];
  red[threadIdx.x] = s;
  __syncthreads();
  for (int o = TPB / 2; o > 0; o >>= 1) {
    if (threadIdx.x < o) red[threadIdx.x] += red[threadIdx.x + o];
    __syncthreads();
  }
  if (threadIdx.x == 0) out[0] = fmaxf(red[0] / Bf, 0.f);
}

extern "C" void kernel_launch(void* const* d_in, const int* in_sizes, int n_in,
                              void* d_out, int out_size, void* d_ws, size_t ws_size,
                              hipStream_t stream) {
  (void)n_in; (void)out_size; (void)ws_size;
  const float* logits  = (const float*)d_in[0];
  const int*   targets = (const int*)d_in[1];
  const float* sim     = (const float*)d_in[2];
  const int Bn = in_sizes[1];               // 524288

  float* wsf   = (float*)d_ws;
  float* Sv    = wsf;                       // [0,170)
  float* Ev    = wsf + 256;                 // [256,426)
  float* parts = wsf + 512;                 // [512,512+NBLK)

  sll_prep<<<(CC + (TPB / 32) - 1) / (TPB / 32), TPB, 0, stream>>>(sim, Sv, Ev);
  const int rpb = Bn / NBLK;                // 512 (B divisible by NBLK)
  sll_main<<<NBLK, TPB, 0, stream>>>(logits, targets, sim, Sv, Ev, parts, rpb);
  sll_finish<<<1, TPB, 0, stream>>>(parts, NBLK, (float*)d_out, (float)Bn);
}